// Leg_fit_4252017623101
// MI455X (gfx1250) — compile-verified
//
#include <hip/hip_runtime.h>
#include <math.h>

#define T_DIM 512
#define N_DIM 384
#define PAD   255   // (K-1)//2 for K=512 SAME correlation

typedef float v2f __attribute__((ext_vector_type(2)));
typedef float v8f __attribute__((ext_vector_type(8)));

// ---------------------------------------------------------------------------
// Kernel 1: compute dopa[t] (SAME corr of z0[:,idx] with reward kernel r_k),
// then g[s] = sum_t dopa[t] * fc[s-t+255].  Also emit Wp/Wm lookup tables.
// Single block of 512 threads; O(T^2) scalar work (~0.5 MFLOP) — negligible.
// ---------------------------------------------------------------------------
__global__ void prep_kernel(const float* __restrict__ z,
                            const float* __restrict__ wmax_p,
                            const int*   __restrict__ idx_p,
                            float* __restrict__ g,
                            float* __restrict__ Wp,
                            float* __restrict__ Wm)
{
    __shared__ float rk[T_DIM];
    __shared__ float fcs[T_DIM];
    __shared__ float dop[T_DIM];

    const int t = threadIdx.x;           // 0..511
    const float wmax = wmax_p[0];
    const int idx = idx_p[0];

    // reward kernel: t_arr[k] = (T - k)/scale
    {
        float tt = (float)(T_DIM - t) / 500.0f;
        rk[t] = 1.379f * tt / 0.2f * __expf(1.0f - tt / 0.2f)
              - 0.27f  * tt / 1.0f * __expf(1.0f - tt / 1.0f);
        float r = (float)t;
        fcs[t] = r / 400.0f * __expf(r / 400.0f);
        Wp[t]  = 0.01f * wmax * __expf(-r / 30.0f);
        Wm[t]  = 0.01f * wmax * 1.05f * __expf(r / 30.0f);
    }
    __syncthreads();

    // dopa[t] = sum_k z0[t+k-255, idx] * rk[k]   (valid source indices only)
    {
        float acc = 0.0f;
        #pragma unroll 4
        for (int k = 0; k < T_DIM; ++k) {
            int s = t + k - PAD;
            if ((unsigned)s < (unsigned)T_DIM)
                acc = fmaf(z[s * N_DIM + idx], rk[k], acc);
        }
        dop[t] = acc;
    }
    __syncthreads();

    // g[s] = sum_t dopa[t] * fc[s - t + 255]   (valid fc indices only)
    {
        float acc = 0.0f;
        #pragma unroll 4
        for (int tt = 0; tt < T_DIM; ++tt) {
            int kk = t - tt + PAD;
            if ((unsigned)kk < (unsigned)T_DIM)
                acc = fmaf(dop[tt], fcs[kk], acc);
        }
        g[t] = acc;
    }
}

// ---------------------------------------------------------------------------
// Kernel 2: H[a,b] = g[b]*Wp[a-b+255] + g[a]*Wm[b-a+255]  (banded validity)
// 512x512 elementwise, trivially parallel.
// ---------------------------------------------------------------------------
__global__ void build_h_kernel(const float* __restrict__ g,
                               const float* __restrict__ Wp,
                               const float* __restrict__ Wm,
                               float* __restrict__ H)
{
    int e = blockIdx.x * blockDim.x + threadIdx.x;   // 0 .. 512*512-1
    int a = e / T_DIM;
    int b = e - a * T_DIM;
    float h = 0.0f;
    int d1 = a - b + PAD;
    if ((unsigned)d1 < (unsigned)T_DIM) h = fmaf(g[b], Wp[d1], h);
    int d2 = b - a + PAD;
    if ((unsigned)d2 < (unsigned)T_DIM) h = fmaf(g[a], Wm[d2], h);
    H[e] = h;
}

// ---------------------------------------------------------------------------
// f32 WMMA GEMM:  C(MxN) = alpha * opA(A) * B.
// One wave computes a 16x64 output strip: 4 independent 16x16 accumulators
// -> 4 parallel WMMA dependency chains through the XDL pipe, and the A
// fragment is loaded once per k-step and reused 4x.
//
// V_WMMA_F32_16X16X4_F32 fragment layout (ISA 7.12.2, wave32):
//   A (16x4): lane<16 -> M=lane, K = k0+{0,1};  lane>=16 -> M=lane-16, K = k0+{2,3}
//   B (4x16): vgpr v  -> row K=k0+v (lanes 0-15), K=k0+v+2 (lanes 16-31), N=lane&15
//   C/D 16x16: vgpr p -> M = p + 8*(lane>=16), N = lane&15
// EXEC all-ones: blockDim = 32, no divergence.
// ---------------------------------------------------------------------------
template <bool TRANS_A>
__global__ void wmma_gemm_f32(const float* __restrict__ A,
                              const float* __restrict__ B,
                              float* __restrict__ C,
                              int K, int lda, int ldb, int ldc, float alpha)
{
    const int lane = threadIdx.x & 31;
    const int half = lane >> 4;       // 0 or 1
    const int r    = lane & 15;
    const int tileM  = blockIdx.x * 16;
    const int tileN0 = blockIdx.y * 64;

    v8f acc0 = {}, acc1 = {}, acc2 = {}, acc3 = {};

    for (int k0 = 0; k0 < K; k0 += 4) {
        const int ka = k0 + 2 * half;

        // Prefetch B rows 8 k-steps ahead (lowers to global_prefetch_b8).
        if (k0 + 8 < K) {
            __builtin_prefetch(&B[(ka + 8) * ldb + tileN0 + r], 0, 1);
            __builtin_prefetch(&B[(ka + 9) * ldb + tileN0 + r], 0, 1);
        }

        v2f a;
        if (TRANS_A) {
            // A[m][k] stored as A[k*lda + m]
            a.x = A[(ka + 0) * lda + (tileM + r)];
            a.y = A[(ka + 1) * lda + (tileM + r)];
        } else {
            a.x = A[(tileM + r) * lda + (ka + 0)];
            a.y = A[(tileM + r) * lda + (ka + 1)];
        }

        v2f b0, b1, b2, b3;
        b0.x = B[(ka + 0) * ldb + tileN0 +  0 + r];
        b0.y = B[(ka + 1) * ldb + tileN0 +  0 + r];
        b1.x = B[(ka + 0) * ldb + tileN0 + 16 + r];
        b1.y = B[(ka + 1) * ldb + tileN0 + 16 + r];
        b2.x = B[(ka + 0) * ldb + tileN0 + 32 + r];
        b2.y = B[(ka + 1) * ldb + tileN0 + 32 + r];
        b3.x = B[(ka + 0) * ldb + tileN0 + 48 + r];
        b3.y = B[(ka + 1) * ldb + tileN0 + 48 + r];

        acc0 = __builtin_amdgcn_wmma_f32_16x16x4_f32(false, a, false, b0,
                                                     (short)0, acc0, false, false);
        acc1 = __builtin_amdgcn_wmma_f32_16x16x4_f32(false, a, false, b1,
                                                     (short)0, acc1, false, false);
        acc2 = __builtin_amdgcn_wmma_f32_16x16x4_f32(false, a, false, b2,
                                                     (short)0, acc2, false, false);
        acc3 = __builtin_amdgcn_wmma_f32_16x16x4_f32(false, a, false, b3,
                                                     (short)0, acc3, false, false);
    }

    const int rowBase = (tileM + 8 * half) * ldc + tileN0 + r;
    #pragma unroll
    for (int p = 0; p < 8; ++p) {
        C[rowBase + p * ldc +  0] = alpha * acc0[p];
        C[rowBase + p * ldc + 16] = alpha * acc1[p];
        C[rowBase + p * ldc + 32] = alpha * acc2[p];
        C[rowBase + p * ldc + 48] = alpha * acc3[p];
    }
}

// ---------------------------------------------------------------------------
// Launch:   out = -(z0^T * H * z0)
//   1) prep:    g, Wp, Wm
//   2) build H  (512x512)
//   3) Y  = H * z0            (512x512 x 512x384)
//   4) out = -(z0^T * Y)      (384x512 x 512x384)
// ---------------------------------------------------------------------------
extern "C" void kernel_launch(void* const* d_in, const int* in_sizes, int n_in,
                              void* d_out, int out_size, void* d_ws, size_t ws_size,
                              hipStream_t stream)
{
    (void)in_sizes; (void)n_in; (void)out_size; (void)ws_size;

    const float* z      = (const float*)d_in[0];   // [1, 512, 384] f32
    const float* wmax   = (const float*)d_in[1];   // scalar f32
    const int*   idx_cn = (const int*)  d_in[2];   // scalar int

    float* ws = (float*)d_ws;
    float* g  = ws;                      // 512
    float* Wp = ws + T_DIM;              // 512
    float* Wm = ws + 2 * T_DIM;          // 512
    float* H  = ws + 3 * T_DIM;          // 512*512
    float* Y  = H + T_DIM * T_DIM;       // 512*384

    // 1) dopa / g / Wp / Wm
    prep_kernel<<<1, T_DIM, 0, stream>>>(z, wmax, idx_cn, g, Wp, Wm);

    // 2) H[a,b]
    build_h_kernel<<<(T_DIM * T_DIM) / 256, 256, 0, stream>>>(g, Wp, Wm, H);

    // 3) Y = H(512x512) * z0(512x384)
    {
        dim3 grid(T_DIM / 16, N_DIM / 64);   // 32 x 6 strips
        wmma_gemm_f32<false><<<grid, 32, 0, stream>>>(
            H, z, Y, /*K=*/T_DIM, /*lda=*/T_DIM, /*ldb=*/N_DIM, /*ldc=*/N_DIM,
            /*alpha=*/1.0f);
    }

    // 4) out = -(z0^T(384x512) * Y(512x384))
    {
        dim3 grid(N_DIM / 16, N_DIM / 64);   // 24 x 6 strips
        wmma_gemm_f32<true><<<grid, 32, 0, stream>>>(
            z, Y, (float*)d_out, /*K=*/T_DIM, /*lda=*/N_DIM, /*ldb=*/N_DIM,
            /*ldc=*/N_DIM, /*alpha=*/-1.0f);
    }
}